// SnakeLoss_3770981286577
// MI455X (gfx1250) — compile-verified
//
#include <hip/hip_runtime.h>
#include <hip/hip_bf16.h>
#include <math.h>

typedef __attribute__((ext_vector_type(2))) float v2f;
typedef __attribute__((ext_vector_type(8))) float v8f;

#define B_SZ 64
#define N_SZ 1024
#define WAVES_PER_WG 4
#define WG_THREADS 128
#define ROWS_PER_WG 64                       // 4 waves x 16 rows
#define WGS_PER_BATCH (N_SZ / ROWS_PER_WG)   // 16
#define NSLICE 8                             // one slice per half-wave
#define SLICE_W 1040                         // used idx range [1,1039]; 1040%64==16
                                             //  -> adjacent half-slices bank-disjoint

// ---------------------------------------------------------------------------
// Zero the per-(batch, shift) accumulation buffer.
// ---------------------------------------------------------------------------
__global__ void snake_zero(float* __restrict__ ws, int n) {
    int i = blockIdx.x * blockDim.x + threadIdx.x;
    if (i < n) ws[i] = 0.0f;
}

// ---------------------------------------------------------------------------
// Core: squared distances via V_WMMA_F32_16X16X4_F32 with augmented vectors,
// single-op clamp + raw v_sqrt_f32, conflict-free ds_add_f32 into half-wave-
// private LDS slices, then global atomic merge into shift_sums[b][s].
// ---------------------------------------------------------------------------
__global__ __launch_bounds__(WG_THREADS)
void snake_shift_sums(const float* __restrict__ x,
                      const float* __restrict__ tgt,
                      float* __restrict__ shift_sums) {
    __shared__ float bins[NSLICE * SLICE_W];   // 32.5 KB

    const int tid  = threadIdx.x;
    const int wave = tid >> 5;
    const int lane = tid & 31;
    const int half = lane >> 4;    // 0: lanes 0-15, 1: lanes 16-31
    const int lm   = lane & 15;

    const int b  = blockIdx.x / WGS_PER_BATCH;
    const int rb = blockIdx.x % WGS_PER_BATCH;
    const int j0 = rb * ROWS_PER_WG + wave * 16;  // this wave's 16 rows

    for (int i = tid; i < NSLICE * SLICE_W; i += WG_THREADS) bins[i] = 0.0f;
    __syncthreads();

    const float2* xp = (const float2*)(x   + (size_t)b * N_SZ * 2);
    const float2* tp = (const float2*)(tgt + (size_t)b * N_SZ * 2);

    // A fragment (16x4, f32): lane half 0 carries K={0,1} = (x0, x1),
    // lane half 1 carries K={2,3} = (|x|^2, 1).   Row M = j0 + lane%16.
    float2 xv = xp[j0 + lm];
    v2f a;
    if (half == 0) { a.x = xv.x;                      a.y = xv.y;  }
    else           { a.x = xv.x * xv.x + xv.y * xv.y; a.y = 1.0f;  }

    // Half-wave-private slice; in-slice index drops the constant j0:
    //   i = r + 8*half - lm + N - k0   in [1, 1039];  s = (i + j0) mod N.
    float* mybins = &bins[(wave * 2 + half) * SLICE_W];
    const int lbase = 8 * half - lm + N_SZ;

    #pragma unroll 2
    for (int kt = 0; kt < N_SZ / 16; ++kt) {
        const int k0 = kt * 16;
        // B fragment (4x16, f32): col N = k0 + lane%16; half 0 carries
        // K={0,1} = (-2*t0, -2*t1), half 1 carries K={2,3} = (1, |t|^2).
        float2 tv = tp[k0 + lm];
        v2f bf;
        if (half == 0) { bf.x = -2.0f * tv.x; bf.y = -2.0f * tv.y;          }
        else           { bf.x = 1.0f;         bf.y = tv.x*tv.x + tv.y*tv.y; }

        v8f c = {0.f, 0.f, 0.f, 0.f, 0.f, 0.f, 0.f, 0.f};
        // D = A x B : 16x16 tile of squared distances
        c = __builtin_amdgcn_wmma_f32_16x16x4_f32(false, a, false, bf,
                                                  (short)0, c, false, false);

        const int ibase = lbase - k0;
        #pragma unroll
        for (int r = 0; r < 8; ++r) {
            float v;
            // single-instruction clamp to [0, inf); max_num drops NaN -> 0
            asm("v_max_num_f32 %0, 0, %1" : "=v"(v) : "v"(c[r]));
            float d = __builtin_amdgcn_sqrtf(v);     // raw v_sqrt_f32
            atomicAdd(&mybins[ibase + r], d);        // ds_add_f32, 32 banks
        }
    }
    __syncthreads();

    // Merge the 8 half-wave slices back to shifts: written index i satisfies
    // s = (i + j0_w) mod N, i in [1,1039] -> i1 = (s - j0_w) & (N-1), and
    // additionally i1 + N when i1 <= SLICE_W - N - 1.
    for (int s = tid; s < N_SZ; s += WG_THREADS) {
        float v = 0.0f;
        #pragma unroll
        for (int h = 0; h < NSLICE; ++h) {
            const int j0w = rb * ROWS_PER_WG + (h >> 1) * 16;
            const int i1  = (s - j0w) & (N_SZ - 1);
            v += bins[h * SLICE_W + i1];
            if (i1 <= SLICE_W - N_SZ - 1)
                v += bins[h * SLICE_W + i1 + N_SZ];
        }
        atomicAdd(&shift_sums[(size_t)b * N_SZ + s], v);  // global_atomic_add_f32
    }
}

// ---------------------------------------------------------------------------
// Per-batch min over the 1024 shift sums (scaled by 1/N).
// ---------------------------------------------------------------------------
__global__ __launch_bounds__(256)
void snake_min(const float* __restrict__ shift_sums, float* __restrict__ mins) {
    __shared__ float red[256];
    const int b = blockIdx.x;
    float m = 3.402823466e38f;
    for (int s = threadIdx.x; s < N_SZ; s += 256)
        m = fminf(m, shift_sums[(size_t)b * N_SZ + s]);
    red[threadIdx.x] = m;
    __syncthreads();
    for (int off = 128; off > 0; off >>= 1) {
        if (threadIdx.x < off)
            red[threadIdx.x] = fminf(red[threadIdx.x], red[threadIdx.x + off]);
        __syncthreads();
    }
    if (threadIdx.x == 0) mins[b] = red[0] * (1.0f / (float)N_SZ);
}

// ---------------------------------------------------------------------------
// Mean over batches -> scalar output.
// ---------------------------------------------------------------------------
__global__ __launch_bounds__(64)
void snake_final(const float* __restrict__ mins, float* __restrict__ out) {
    __shared__ float red[64];
    red[threadIdx.x] = mins[threadIdx.x];
    __syncthreads();
    for (int off = 32; off > 0; off >>= 1) {
        if (threadIdx.x < off) red[threadIdx.x] += red[threadIdx.x + off];
        __syncthreads();
    }
    if (threadIdx.x == 0) out[0] = red[0] * (1.0f / (float)B_SZ);
}

extern "C" void kernel_launch(void* const* d_in, const int* in_sizes, int n_in,
                              void* d_out, int out_size, void* d_ws, size_t ws_size,
                              hipStream_t stream) {
    const float* x = (const float*)d_in[0];   // [64, 1024, 2]
    const float* t = (const float*)d_in[1];   // [64, 1024, 2]
    float* ws         = (float*)d_ws;
    float* shift_sums = ws;                   // B*N floats (256 KB)
    float* mins       = ws + B_SZ * N_SZ;     // B floats

    const int nz = B_SZ * N_SZ;
    snake_zero<<<(nz + 255) / 256, 256, 0, stream>>>(shift_sums, nz);
    snake_shift_sums<<<B_SZ * WGS_PER_BATCH, WG_THREADS, 0, stream>>>(x, t, shift_sums);
    snake_min<<<B_SZ, 256, 0, stream>>>(shift_sums, mins);
    snake_final<<<1, 64, 0, stream>>>(mins, (float*)d_out);
}